// MergeLayer_4956392259721
// MI455X (gfx1250) — compile-verified
//
#include <hip/hip_runtime.h>
#include <hip/hip_bf16.h>
#include <math.h>

// ---------------------------------------------------------------------------
// GAT layer for MI455X (gfx1250, wave32):
//   z_dst = h0 @ W_dst            -> WMMA f32 16x16x4 GEMM
//   s_src = h1 @ a_w[:D], s_dst = z_dst @ a_w[D:]
//   e = leaky_relu(s_src[src]+s_dst[dst]); segment softmax over dst
//   out = z_dst/deg + sum(alpha * h1[src])   (0 where deg==0)
// ---------------------------------------------------------------------------

#define D_IN 128
#define D    64
#define NEG_SLOPE 0.01f

typedef __attribute__((ext_vector_type(2))) float v2f;
typedef __attribute__((ext_vector_type(8))) float v8f;

// ---------------- init: zero out / denom / deg, m = -inf --------------------
__global__ void init_kernel(float* out, float* m, float* denom, int* deg, int N) {
    int idx = blockIdx.x * blockDim.x + threadIdx.x;
    if (idx < N * D) out[idx] = 0.0f;
    if (idx < N) {
        m[idx]     = -INFINITY;
        denom[idx] = 0.0f;
        deg[idx]   = 0;
    }
}

// ---------------- GEMM: z = h0 @ W  (N x 128) @ (128 x 64) -----------------
// block = 128 threads (4 waves). Block owns 16 rows; wave w owns cols [16w,16w+16).
__global__ void __launch_bounds__(128)
gemm_zdst_kernel(const float* __restrict__ h0, const float* __restrict__ W,
                 float* __restrict__ z, int N) {
    __shared__ float ldsA[16 * D_IN];   // 8 KB tile of h0

    const int tid  = threadIdx.x;
    const int wid  = tid >> 5;          // wave id 0..3
    const int lane = tid & 31;
    const int r0   = blockIdx.x * 16;   // first row of tile

    // hint: W (32 KB) is hot for every block
    __builtin_prefetch(W, 0, 3);

    // cooperative LDS fill: 16 rows x 128 cols, 128 threads -> 16 floats each
#pragma unroll
    for (int i = 0; i < 16; ++i) {
        int row = r0 + i;
        ldsA[i * D_IN + tid] = (row < N) ? h0[(size_t)row * D_IN + tid] : 0.0f;
    }
    __syncthreads();

    const int mrow = lane & 15;          // M index within tile
    const int khalf = (lane >> 4) * 2;   // K sub-offset (0 or 2)
    const int n0 = wid * 16;             // column tile base
    const int ncol = n0 + mrow;          // this lane's B/C column

    v8f c = {};                          // 16x16 f32 accumulator (8 VGPRs)

#pragma unroll
    for (int k = 0; k < D_IN; k += 4) {
        const int kk = k + khalf;
        // A fragment 16x4: lane holds A[mrow][kk], A[mrow][kk+1] (contig -> ds_load_b64)
        v2f a;
        a.x = ldsA[mrow * D_IN + kk];
        a.y = ldsA[mrow * D_IN + kk + 1];
        // B fragment 4x16: lane holds W[kk][ncol], W[kk+1][ncol]
        v2f b;
        b.x = W[(size_t)kk * D + ncol];
        b.y = W[(size_t)(kk + 1) * D + ncol];
        c = __builtin_amdgcn_wmma_f32_16x16x4_f32(
                /*neg_a=*/false, a, /*neg_b=*/false, b,
                /*c_mod=*/(short)0, c, /*reuse_a=*/false, /*reuse_b=*/false);
    }

    // C/D layout: VGPR r -> row (r + 8*(lane/16)), col = lane%16
#pragma unroll
    for (int r = 0; r < 8; ++r) {
        int row = r0 + r + ((lane >> 4) << 3);
        if (row < N) z[(size_t)row * D + ncol] = c[r];
    }
}

// ---------------- per-node scores: wave per node ---------------------------
__global__ void __launch_bounds__(256)
scores_kernel(const float* __restrict__ h1, const float* __restrict__ z,
              const float* __restrict__ a_w, float* __restrict__ s_src,
              float* __restrict__ s_dst, int N) {
    const int lane = threadIdx.x & 31;
    const int node = blockIdx.x * 8 + (threadIdx.x >> 5);
    if (node >= N) return;

    float ps = h1[(size_t)node * D + lane]      * a_w[lane]
             + h1[(size_t)node * D + lane + 32] * a_w[lane + 32];
    float pd = z[(size_t)node * D + lane]       * a_w[D + lane]
             + z[(size_t)node * D + lane + 32]  * a_w[D + lane + 32];
#pragma unroll
    for (int off = 16; off >= 1; off >>= 1) {
        ps += __shfl_xor(ps, off, 32);
        pd += __shfl_xor(pd, off, 32);
    }
    if (lane == 0) {
        s_src[node] = ps;
        s_dst[node] = pd;
    }
}

// float atomic max via sign-split (m initialized to -inf)
__device__ inline void atomicMaxF32(float* addr, float val) {
    if (val >= 0.0f) atomicMax((int*)addr, __float_as_int(val));
    else             atomicMin((unsigned int*)addr, __float_as_uint(val));
}

// ---------------- edge pass 1: e, segment max, degree ----------------------
__global__ void __launch_bounds__(256)
edge_score_kernel(const float* __restrict__ s_src, const float* __restrict__ s_dst,
                  const int* __restrict__ src, const int* __restrict__ dst,
                  float* __restrict__ e_buf, float* __restrict__ m,
                  int* __restrict__ deg, int E) {
    int t = blockIdx.x * blockDim.x + threadIdx.x;
    if (t >= E) return;
    int d = dst[t];
    float x = s_src[src[t]] + s_dst[d];
    float e = (x > 0.0f) ? x : NEG_SLOPE * x;   // leaky_relu
    e_buf[t] = e;
    atomicMaxF32(&m[d], e);
    atomicAdd(&deg[d], 1);
}

// ---------------- edge pass 2: exp + denom ---------------------------------
__global__ void __launch_bounds__(256)
edge_exp_kernel(const int* __restrict__ dst, const float* __restrict__ m,
                float* __restrict__ e_buf, float* __restrict__ denom, int E) {
    int t = blockIdx.x * blockDim.x + threadIdx.x;
    if (t >= E) return;
    int d = dst[t];
    float mv = m[d];                 // finite: this edge contributed to the max
    float ex = __expf(e_buf[t] - mv);
    e_buf[t] = ex;                   // overwrite e with exp(e - m)
    atomicAdd(&denom[d], ex);
}

// ---------------- edge pass 3: weighted scatter (wave per edge) ------------
__global__ void __launch_bounds__(256)
edge_agg_kernel(const float* __restrict__ h1, const float* __restrict__ e_buf,
                const float* __restrict__ denom, const int* __restrict__ src,
                const int* __restrict__ dst, float* __restrict__ out, int E) {
    const int lane = threadIdx.x & 31;
    const int t = blockIdx.x * 8 + (threadIdx.x >> 5);
    if (t >= E) return;
    int s = src[t], d = dst[t];
    float alpha = e_buf[t] / denom[d];
    // h1 (12.8 MB) and out (12.8 MB) are L2-resident -> L2 atomic units do the RMW
    atomicAdd(&out[(size_t)d * D + lane],      alpha * h1[(size_t)s * D + lane]);
    atomicAdd(&out[(size_t)d * D + lane + 32], alpha * h1[(size_t)s * D + lane + 32]);
}

// ---------------- finalize: out = z/deg + agg (0 if deg==0) ----------------
__global__ void __launch_bounds__(256)
finalize_kernel(const float* __restrict__ z, const int* __restrict__ deg,
                float* __restrict__ out, int N) {
    int idx = blockIdx.x * blockDim.x + threadIdx.x;
    if (idx >= N * D) return;
    int row = idx / D;
    int dg = deg[row];
    out[idx] = (dg > 0) ? (z[idx] / (float)dg + out[idx]) : 0.0f;
}

// ---------------------------------------------------------------------------
extern "C" void kernel_launch(void* const* d_in, const int* in_sizes, int n_in,
                              void* d_out, int out_size, void* d_ws, size_t ws_size,
                              hipStream_t stream) {
    const float* h0  = (const float*)d_in[0];   // [N, 128]
    const float* h1  = (const float*)d_in[1];   // [N, 64]
    const float* W   = (const float*)d_in[2];   // [128, 64]
    const float* a_w = (const float*)d_in[3];   // [128]
    const int*   src = (const int*)d_in[4];     // [E]
    const int*   dst = (const int*)d_in[5];     // [E]
    float* out = (float*)d_out;                 // [N, 64]

    const int N = in_sizes[1] / D;
    const int E = in_sizes[4];

    // workspace layout
    float* z     = (float*)d_ws;                // N*64
    float* ssrc  = z + (size_t)N * D;           // N
    float* sdst  = ssrc + N;                    // N
    float* m     = sdst + N;                    // N
    float* denom = m + N;                       // N
    float* e_buf = denom + N;                   // E
    int*   deg   = (int*)(e_buf + E);           // N

    const int T = 256;

    init_kernel<<<(N * D + T - 1) / T, T, 0, stream>>>(out, m, denom, deg, N);

    gemm_zdst_kernel<<<(N + 15) / 16, 128, 0, stream>>>(h0, W, z, N);

    scores_kernel<<<(N + 7) / 8, T, 0, stream>>>(h1, z, a_w, ssrc, sdst, N);

    edge_score_kernel<<<(E + T - 1) / T, T, 0, stream>>>(ssrc, sdst, src, dst,
                                                         e_buf, m, deg, E);

    edge_exp_kernel<<<(E + T - 1) / T, T, 0, stream>>>(dst, m, e_buf, denom, E);

    edge_agg_kernel<<<(E + 7) / 8, T, 0, stream>>>(h1, e_buf, denom, src, dst, out, E);

    finalize_kernel<<<(N * D + T - 1) / T, T, 0, stream>>>(z, deg, out, N);
}